// PlanarLoss_44169443672384
// MI455X (gfx1250) — compile-verified
//
#include <hip/hip_runtime.h>

typedef __attribute__((ext_vector_type(2))) float v2f;
typedef __attribute__((ext_vector_type(8))) float v8f;

#define NW   8      // waves per block (wave32)
#define ROWS 32     // rows of the distance matrix handled per block
#define TK   8      // neighbors kept (k in the reference)

__global__ void PlanarLoss_zero_kernel(float* out) { out[0] = 0.0f; }

// Branchless sorted-insert: swaps (d,id) through the ascending list.
// Fully unrolled -> pure v_cndmask chains, no EXEC manipulation.
__device__ __forceinline__ void topk_insert(float (&bd)[TK], int (&bi)[TK],
                                            float d, int id) {
#pragma unroll
  for (int s = 0; s < TK; ++s) {
    const bool  c   = d < bd[s];
    const float nbd = c ? d     : bd[s];
    const int   nbi = c ? id    : bi[s];
    const float nd  = c ? bd[s] : d;
    const int   nid = c ? bi[s] : id;
    bd[s] = nbd; bi[s] = nbi; d = nd; id = nid;
  }
}

__global__ __launch_bounds__(256) void PlanarLoss_knn_kernel(
    const float* __restrict__ means,
    const float* __restrict__ normals,
    float* __restrict__ out,
    int N, float scale)
{
  __shared__ float lds_tile[NW][ROWS][16];  // staged WMMA D tiles (Gram values)
  __shared__ float lds_sqj[NW][16];         // |p_j|^2 for current column tile
  __shared__ float cand_d[NW][ROWS][TK];    // per-wave partial top-k (d^2)
  __shared__ int   cand_i[NW][ROWS][TK];    // per-wave partial top-k (index)

  const int  tid  = threadIdx.x;
  const int  lane = tid & 31;
  const int  wid  = tid >> 5;
  const int  row0 = blockIdx.x * ROWS;
  const bool lo   = lane < 16;
  const int  m    = lane & 15;

  // A matrices for rows [row0, row0+16) and [row0+16, row0+32).
  // f32 16x4 A layout: lanes 0-15 -> VGPR0=K0, VGPR1=K1; lanes 16-31 -> K2, K3.
  // Point (x,y,z) padded with K3 = 0.
  const float* pr0 = means + (size_t)(row0 + m) * 3;
  const float* pr1 = means + (size_t)(row0 + 16 + m) * 3;
  v2f a0 = { lo ? pr0[0] : pr0[2], lo ? pr0[1] : 0.0f };
  v2f a1 = { lo ? pr1[0] : pr1[2], lo ? pr1[1] : 0.0f };

  // Each lane owns one of the 32 rows for the selection phase.
  const int   i   = row0 + lane;
  const float mix = means[(size_t)i * 3 + 0];
  const float miy = means[(size_t)i * 3 + 1];
  const float miz = means[(size_t)i * 3 + 2];
  const float sqi = mix * mix + miy * miy + miz * miz;

  float bd[TK]; int bi[TK];
#pragma unroll
  for (int s = 0; s < TK; ++s) { bd[s] = 3.4e38f; bi[s] = -1; }

  const int ntiles = N >> 4;
  for (int tj = wid; tj < ntiles; tj += NW) {
    const int col0 = tj << 4;

    // B matrix (4x16) for this column tile: same per-lane packing as A.
    // Both half-waves load the SAME 16 points, so the sq store below is
    // an unconditional duplicate write of identical values (no divergence).
    const float* pc = means + (size_t)(col0 + m) * 3;
    const float px = pc[0], py = pc[1], pz = pc[2];
    v2f b = { lo ? px : pz, lo ? py : 0.0f };
    lds_sqj[wid][m] = px * px + py * py + pz * pz;

    // G = A * B (Gram matrix tile), exact f32 accumulation.
    v8f c = {};
    v8f g0 = __builtin_amdgcn_wmma_f32_16x16x4_f32(false, a0, false, b, (short)0, c, false, false);
    v8f g1 = __builtin_amdgcn_wmma_f32_16x16x4_f32(false, a1, false, b, (short)0, c, false, false);

    // D layout: lane L, VGPR r -> row r + (L<16 ? 0 : 8), col L%16.
    const int rb = lo ? 0 : 8;
#pragma unroll
    for (int r = 0; r < 8; ++r) lds_tile[wid][rb + r][m] = g0[r];
#pragma unroll
    for (int r = 0; r < 8; ++r) lds_tile[wid][16 + rb + r][m] = g1[r];
    __syncthreads();

    // Selection: lane L scans its row's 16 new candidates.
    // sqrt is monotone -> rank on d^2 directly (no sqrt needed).
    const float* rowp = &lds_tile[wid][lane][0];
    for (int n = 0; n < 16; ++n) {
      const int   j  = col0 + n;
      const float d2 = sqi + lds_sqj[wid][n] - 2.0f * rowp[n];
      if (j != i && d2 < bd[TK - 1]) {   // rare-path guard (self excluded)
        topk_insert(bd, bi, d2, j);
      }
    }
    __syncthreads();                     // tile buffer reused next iter
  }

  // Merge the 8 per-wave partial top-8 lists per row.
#pragma unroll
  for (int s = 0; s < TK; ++s) { cand_d[wid][lane][s] = bd[s]; cand_i[wid][lane][s] = bi[s]; }
  __syncthreads();

  if (tid < ROWS) {
    float fd[TK]; int fi[TK];
#pragma unroll
    for (int s = 0; s < TK; ++s) { fd[s] = 3.4e38f; fi[s] = -1; }
    for (int w = 0; w < NW; ++w) {
#pragma unroll
      for (int s0 = 0; s0 < TK; ++s0) {
        const float d  = cand_d[w][tid][s0];
        const int   id = cand_i[w][tid][s0];
        if (d < fd[TK - 1]) topk_insert(fd, fi, d, id);
      }
    }

    // Plane-distance reduction for this row.
    const int   ii = row0 + tid;
    const float nx = normals[(size_t)ii * 3 + 0];
    const float ny = normals[(size_t)ii * 3 + 1];
    const float nz = normals[(size_t)ii * 3 + 2];
    const float ax = means[(size_t)ii * 3 + 0];
    const float ay = means[(size_t)ii * 3 + 1];
    const float az = means[(size_t)ii * 3 + 2];
    float acc = 0.0f;
#pragma unroll
    for (int s = 0; s < TK; ++s) {
      const int j = fi[s];
      if (j >= 0) {
        const float dx = means[(size_t)j * 3 + 0] - ax;
        const float dy = means[(size_t)j * 3 + 1] - ay;
        const float dz = means[(size_t)j * 3 + 2] - az;
        acc += fabsf(dx * nx + dy * ny + dz * nz);
      }
    }
    atomicAdd(out, acc * scale);
  }
}

extern "C" void kernel_launch(void* const* d_in, const int* in_sizes, int n_in,
                              void* d_out, int out_size, void* d_ws, size_t ws_size,
                              hipStream_t stream) {
  const float* means   = (const float*)d_in[0];
  const float* normals = (const float*)d_in[1];
  float* out = (float*)d_out;

  const int   N     = in_sizes[0] / 3;                 // 8192
  const float scale = 1.0f / ((float)N * (float)TK);   // mean over N*k terms

  PlanarLoss_zero_kernel<<<1, 1, 0, stream>>>(out);
  PlanarLoss_knn_kernel<<<N / ROWS, 256, 0, stream>>>(means, normals, out, N, scale);
}